// SSKernelNPLR_2619930050870
// MI455X (gfx1250) — compile-verified
//
#include <hip/hip_runtime.h>
#include <math.h>

typedef float v2f __attribute__((ext_vector_type(2)));
typedef float v8f __attribute__((ext_vector_type(8)));

#define HH    512
#define NN    32
#define LLEN  1024
#define LF    513           // L/2 + 1
#define KTOT  1026          // LF (cos rows) + LF (sin rows)
#define KPAD  1028          // padded to multiple of 4 for 16x16x4 WMMA
#define KPAIR (KPAD/2)      // 514 row-pairs

// Workspace layout (floats):
//   wsA : [KPAIR][HH][2]    pair-interleaved k_f   (k at [(k>>1)][h][k&1])
//   wsB : [KPAIR][LLEN][2]  pair-interleaved basis (k at [(k>>1)][l][k&1])
#define WSA_FLOATS (KPAD*HH)

// ---------------------------------------------------------------------------
// Stage 1: Cauchy kernel + rank-1 Woodbury correction.  One block per h.
// Stable form 1/(z-w) = (1+omega)/(2(1-omega) - w_dt*(1+omega)) keeps the
// Nyquist bin (omega -> -1) well conditioned.
// ---------------------------------------------------------------------------
__global__ __launch_bounds__(256) void s4_stage1(
    const float* __restrict__ log_dt,      // [H]
    const float* __restrict__ inv_w_real,  // [1][32]
    const float* __restrict__ w_imag,      // [1][32]
    const float* __restrict__ B_ri,        // [1][1][32][2]
    const float* __restrict__ C_ri,        // [1][H][32][2]
    const float* __restrict__ P_ri,        // [1][1][32][2]
    float* __restrict__ wsA)               // [KPAIR][HH][2]
{
  __shared__ float sw2r[NN], sw2i[NN];
  __shared__ float sv00r[NN], sv00i[NN];
  __shared__ float sv01r[NN], sv01i[NN];
  __shared__ float sv10r[NN], sv10i[NN];
  __shared__ float sv11r[NN];

  const int h   = blockIdx.x;
  const int tid = threadIdx.x;
  const float dt = expf(log_dt[h]);

  if (tid < NN) {
    const int n = tid;
    const float wr = -expf(inv_w_real[n]);
    const float wi = w_imag[n];
    sw2r[n] = wr * dt;
    sw2i[n] = wi * dt;
    const float br = B_ri[2*n],            bi = B_ri[2*n+1];
    const float pr = P_ri[2*n],            pi = P_ri[2*n+1];
    const float cr = C_ri[(h*NN + n)*2],   ci = C_ri[(h*NN + n)*2 + 1];
    // v00 = B*C ; v01 = B*conj(P) ; v10 = P*C ; v11 = |P|^2 (real)
    sv00r[n] = br*cr - bi*ci;  sv00i[n] = br*ci + bi*cr;
    sv01r[n] = br*pr + bi*pi;  sv01i[n] = bi*pr - br*pi;
    sv10r[n] = pr*cr - pi*ci;  sv10i[n] = pr*ci + pi*cr;
    sv11r[n] = pr*pr + pi*pi;
  }
  __syncthreads();

  for (int f = tid; f < LF; f += blockDim.x) {
    const float ang = -3.14159265358979323846f * (float)f * (1.0f/512.0f);
    const float owr = cosf(ang), owi = sinf(ang);      // omega_f
    const float Ar  = 2.0f - 2.0f*owr, Ai = -2.0f*owi; // 2(1-omega)
    const float Br  = 1.0f + owr,      Bi = owi;       // (1+omega)

    float t00r=0.f,t00i=0.f,t01r=0.f,t01i=0.f,t10r=0.f,t10i=0.f,t11r=0.f,t11i=0.f;

    #pragma unroll 4
    for (int n = 0; n < NN; ++n) {
      const float w2r = sw2r[n], w2i = sw2i[n];
      {  // lower half: D = A - w2*(1+omega)
        const float Dr = Ar - (w2r*Br - w2i*Bi);
        const float Di = Ai - (w2r*Bi + w2i*Br);
        const float s  = 1.0f / (Dr*Dr + Di*Di);
        const float rr = Dr*s, ri = -Di*s;
        float vr, vi;
        vr = sv00r[n]; vi = sv00i[n];
        t00r += vr*rr - vi*ri; t00i += vr*ri + vi*rr;
        vr = sv01r[n]; vi = sv01i[n];
        t01r += vr*rr - vi*ri; t01i += vr*ri + vi*rr;
        vr = sv10r[n]; vi = sv10i[n];
        t10r += vr*rr - vi*ri; t10i += vr*ri + vi*rr;
        vr = sv11r[n];
        t11r += vr*rr;         t11i += vr*ri;
      }
      {  // upper half: conjugated poles and residues
        const float Dr = Ar - (w2r*Br + w2i*Bi);
        const float Di = Ai - (w2r*Bi - w2i*Br);
        const float s  = 1.0f / (Dr*Dr + Di*Di);
        const float rr = Dr*s, ri = -Di*s;
        float vr, vi;
        vr = sv00r[n]; vi = -sv00i[n];
        t00r += vr*rr - vi*ri; t00i += vr*ri + vi*rr;
        vr = sv01r[n]; vi = -sv01i[n];
        t01r += vr*rr - vi*ri; t01i += vr*ri + vi*rr;
        vr = sv10r[n]; vi = -sv10i[n];
        t10r += vr*rr - vi*ri; t10i += vr*ri + vi*rr;
        vr = sv11r[n];
        t11r += vr*rr;         t11i += vr*ri;
      }
    }

    // k_f = 2*dt*( T00 - u*T01*T10 / (1 + u*T11) ), u = dt*(1+omega)
    const float ur = dt*Br, ui = dt*Bi;
    const float denr = 1.0f + (ur*t11r - ui*t11i);
    const float deni =        (ur*t11i + ui*t11r);
    const float ar = ur*t01r - ui*t01i, ai = ur*t01i + ui*t01r;
    const float nr = ar*t10r - ai*t10i, ni = ar*t10i + ai*t10r;
    const float s  = 1.0f / (denr*denr + deni*deni);
    const float qr = (nr*denr + ni*deni)*s;
    const float qi = (ni*denr - nr*deni)*s;

    const int kRe = f;         // cos-row index
    const int kIm = LF + f;    // sin-row index
    wsA[(kRe >> 1)*(2*HH) + 2*h + (kRe & 1)] = 2.0f*dt*(t00r - qr);
    wsA[(kIm >> 1)*(2*HH) + 2*h + (kIm & 1)] = 2.0f*dt*(t00i - qi);
  }

  // zero the K-padding pair (k = 1026, 1027)
  if (h == 0) {
    for (int i = tid; i < 2*HH; i += blockDim.x)
      wsA[(KPAD - 2)*HH + i] = 0.0f;
  }
}

// ---------------------------------------------------------------------------
// Stage 1b: precompute irfft basis, pair-interleaved [KPAIR][LLEN][2].
//   row k < LF        :  scale_f * cos(2*pi*f*l/L)/L        (f = k)
//   LF <= k < KTOT    : -scale_f * sin(2*pi*f*l/L)/L        (f = k-LF)
//   k >= KTOT         :  0
//   scale_{0,512} = 1, else 2.  Exact integer phase reduction mod L.
// ---------------------------------------------------------------------------
__device__ __forceinline__ float basis_val(int q, int l) {
  if (q >= KTOT) return 0.0f;
  const bool isSin = (q >= LF);
  const int  f = isSin ? (q - LF) : q;
  const float sc = ((f == 0 || f == (LLEN/2)) ? 1.0f : 2.0f) * (1.0f/(float)LLEN);
  const int   t = (f * l) & (LLEN - 1);
  const float ang = (float)t * (2.0f * 3.14159265358979323846f / (float)LLEN);
  return isSin ? (-sc * sinf(ang)) : (sc * cosf(ang));
}

__global__ __launch_bounds__(256) void s4_basis(float* __restrict__ wsB) {
  const int idx = blockIdx.x * 256 + threadIdx.x;   // pair-element index
  if (idx >= KPAIR * LLEN) return;
  const int pair = idx / LLEN;
  const int l    = idx - pair * LLEN;
  v2f b;
  b.x = basis_val(2*pair,     l);
  b.y = basis_val(2*pair + 1, l);
  *reinterpret_cast<v2f*>(wsB + 2*idx) = b;
}

// ---------------------------------------------------------------------------
// Stage 2: out[512][1024] = kf[512][1028] @ basis[1028][1024] via
// V_WMMA_F32_16X16X4_F32.  One wave per 16x16 output tile; each K-step is
// exactly two coalesced b64 loads + one WMMA (A/B fragments follow the ISA
// wave32 layouts: lanes 0-15 K={0,1}, lanes 16-31 K={2,3}).
// ---------------------------------------------------------------------------
__global__ __launch_bounds__(256) void s4_stage2(
    const float* __restrict__ ws,    // wsA | wsB
    float* __restrict__ out)         // [H][L]
{
  const int tid  = threadIdx.x;
  const int wave = tid >> 5;
  const int lane = tid & 31;
  const int half = lane >> 4;
  const int n16  = lane & 15;

  const int tile = blockIdx.x * 8 + wave;   // 2048 tiles total
  const int tm   = tile >> 6;               // 0..31 over H
  const int tn   = tile & 63;               // 0..63 over L
  const int m    = tm*16 + n16;             // A row (h)
  const int l    = tn*16 + n16;             // output column

  const v2f* __restrict__ Ap = reinterpret_cast<const v2f*>(ws);
  const v2f* __restrict__ Bp = reinterpret_cast<const v2f*>(ws + WSA_FLOATS);

  v8f acc = {};
  #pragma unroll 4
  for (int kk = 0; kk < KPAD; kk += 4) {
    const int pair = (kk >> 1) + half;      // rows (kk+2*half, kk+2*half+1)
    const v2f a = Ap[pair*HH   + m];        // b64, coalesced across lanes
    const v2f b = Bp[pair*LLEN + l];        // b64, coalesced across lanes
    acc = __builtin_amdgcn_wmma_f32_16x16x4_f32(
        /*neg_a=*/false, a, /*neg_b=*/false, b,
        /*c_mod=*/(short)0, acc, /*reuse_a=*/false, /*reuse_b=*/false);
  }

  #pragma unroll
  for (int v = 0; v < 8; ++v) {
    const int row = tm*16 + v + 8*half;     // C/D layout: lanes 16-31 hold M+8
    out[row*LLEN + l] = acc[v];
  }
}

// ---------------------------------------------------------------------------
extern "C" void kernel_launch(void* const* d_in, const int* in_sizes, int n_in,
                              void* d_out, int out_size, void* d_ws, size_t ws_size,
                              hipStream_t stream) {
  const float* log_dt     = (const float*)d_in[0];
  const float* inv_w_real = (const float*)d_in[1];
  const float* w_imag     = (const float*)d_in[2];
  const float* B_ri       = (const float*)d_in[3];
  const float* C_ri       = (const float*)d_in[4];
  const float* P_ri       = (const float*)d_in[5];
  // d_in[6] = L (python scalar, fixed at 1024 here)

  float* ws  = (float*)d_ws;                // needs (KPAD*HH + KPAD*LLEN)*4 ~ 6.3 MB
  float* wsA = ws;
  float* wsB = ws + WSA_FLOATS;

  s4_stage1<<<HH, 256, 0, stream>>>(log_dt, inv_w_real, w_imag, B_ri, C_ri, P_ri, wsA);
  const int basisElems  = KPAIR * LLEN;
  const int basisBlocks = (basisElems + 255) / 256;
  s4_basis<<<basisBlocks, 256, 0, stream>>>(wsB);
  s4_stage2<<<256, 256, 0, stream>>>(ws, (float*)d_out);
}